// KNO2d_59210419142753
// MI455X (gfx1250) — compile-verified
//
#include <hip/hip_runtime.h>
#include <math.h>

// ---------------------------------------------------------------------------
// KNO2d on gfx1250 (MI455X).  f32 end-to-end, matmuls via V_WMMA_F32_16X16X4_F32.
// Sizes: B=32, CIN=6, COUT=4, OP=32, MX=MY=12, DEC=6, H=W=192.
// Spectral trick: the 6 Koopman iterations collapse to tiny 288-mode spectral
// algebra; one forward DFT (onto 24x12 modes) and one inverse DFT.
// ---------------------------------------------------------------------------

typedef float v2f __attribute__((ext_vector_type(2)));
typedef float v8f __attribute__((ext_vector_type(8)));

#define TWO_PI_OVER_N (6.283185307179586f / 192.0f)

__device__ __forceinline__ v8f wmma_f32(v2f a, v2f b, v8f c) {
  return __builtin_amdgcn_wmma_f32_16x16x4_f32(false, a, false, b, (short)0, c,
                                               false, false);
}

// ---------------------------------------------------------------------------
// Kernel 0: precompute DFT coefficient tables (global constants, ~18KB each).
//   Fyg[y*24 + n]  : forward rfft along y.  n = 2*ky+p: p=0 -> cos, p=1 -> -sin
//   B2g[k*192 + y] : irfft along y with 1/192.  k = 2*ky+p.
// ---------------------------------------------------------------------------
__global__ __launch_bounds__(256) void k_tables(float* __restrict__ Fyg,
                                                float* __restrict__ B2g) {
  const int i = blockIdx.x * 256 + threadIdx.x;
  if (i >= 4608) return;
  const float invW = 1.0f / 192.0f;
  {
    int y = i / 24, n = i % 24, ky = n >> 1, p = n & 1;
    float s, c;
    sincosf(TWO_PI_OVER_N * (float)((ky * y) % 192), &s, &c);
    Fyg[i] = p ? -s : c;
  }
  {
    int k = i / 192, y = i % 192, ky = k >> 1, p = k & 1;
    float v;
    if (ky == 0) {
      v = p ? 0.f : invW;  // Im of ky=0 bin ignored by irfft
    } else {
      float s, c;
      sincosf(TWO_PI_OVER_N * (float)((ky * y) % 192), &s, &c);
      v = p ? (-2.f * s * invW) : (2.f * c * invW);
    }
    B2g[i] = v;
  }
}

// ---------------------------------------------------------------------------
// Kernel 1: enc 1x1 conv (6->32) + tanh, then DFT along y onto ky=0..11.
// One block per (b, x-row). Output g[b][ch][ky][x] (complex, float2).
// WMMA: G[32ch x 24n] = Y[32ch x 192y] * Fy[192y x 24n], K=192.
// ---------------------------------------------------------------------------
__global__ __launch_bounds__(128) void k_enc_dfty(
    const float* __restrict__ x, const float* __restrict__ enc_w,
    const float* __restrict__ enc_b, const float* __restrict__ Fyg,
    float* __restrict__ g) {
  __shared__ float xin[6][193];
  __shared__ float ylds[32][193];
  __shared__ float FyL[192][33];  // cols 24..32 zero (padding for N tiles)
  __shared__ float wenc[192];
  __shared__ float benc[32];

  const int bid = blockIdx.x;
  const int b = bid / 192, xr = bid % 192;
  const int tid = threadIdx.x;

  for (int i = tid; i < 192; i += 128) wenc[i] = enc_w[i];
  if (tid < 32) benc[tid] = enc_b[tid];
  for (int i = tid; i < 6 * 192; i += 128) {
    int c = i / 192, y = i % 192;
    xin[c][y] = x[((b * 6 + c) * 192 + xr) * 192 + y];
  }
  for (int i = tid; i < 192 * 24; i += 128) FyL[i / 24][i % 24] = Fyg[i];
  for (int i = tid; i < 192 * 9; i += 128) FyL[i / 9][24 + i % 9] = 0.f;
  __syncthreads();

  for (int i = tid; i < 32 * 192; i += 128) {
    int ch = i / 192, y = i % 192;
    float s = benc[ch];
#pragma unroll
    for (int c = 0; c < 6; ++c) s += wenc[ch * 6 + c] * xin[c][y];
    ylds[ch][y] = tanhf(s);
  }
  __syncthreads();

  const int lane = tid & 31, wv = tid >> 5;
  const int mtile = (wv >> 1) * 16;  // channel tile: 0 or 16
  const int ntile = (wv & 1) * 16;   // n = 2*ky + (re/im) tile
  const int h = lane >> 4, l15 = lane & 15;
  const int col = ntile + l15;       // n index, valid if < 24

  v8f acc = {0.f, 0.f, 0.f, 0.f, 0.f, 0.f, 0.f, 0.f};
  for (int k0 = 0; k0 < 192; k0 += 4) {
    v2f a, bf;
    a.x = ylds[mtile + l15][k0 + 2 * h];
    a.y = ylds[mtile + l15][k0 + 2 * h + 1];
    bf.x = FyL[k0 + 2 * h][col];
    bf.y = FyL[k0 + 2 * h + 1][col];
    acc = wmma_f32(a, bf, acc);
  }

  if (col < 24) {
    const int ky = col >> 1, par = col & 1;
#pragma unroll
    for (int i = 0; i < 8; ++i) {
      int ch = mtile + 8 * h + i;
      g[((((b * 32 + ch) * 12 + ky) * 192 + xr) << 1) + par] = acc[i];
    }
  }
}

// ---------------------------------------------------------------------------
// Kernel 2: DFT along x onto kx in {0..11} u {180..191}.
// One block per (b,ch).  S[b][ch][mode], mode = kx_i*12 + ky (24x12).
// ---------------------------------------------------------------------------
__global__ __launch_bounds__(256) void k_dftx(const float* __restrict__ g,
                                              float* __restrict__ S) {
  __shared__ float2 gl[12 * 192];
  __shared__ float2 tw[192];
  const int bc = blockIdx.x;  // b*32 + ch
  const int tid = threadIdx.x;

  for (int i = tid; i < 192; i += 256) {
    float s, c;
    sincosf(-TWO_PI_OVER_N * (float)i, &s, &c);
    tw[i] = make_float2(c, s);  // e^{-2*pi*i*t/192}
  }
  const float2* gp = (const float2*)g + (size_t)bc * 12 * 192;
  for (int i = tid; i < 2304; i += 256) gl[i] = gp[i];
  __syncthreads();

  for (int o = tid; o < 288; o += 256) {
    int kx_i = o / 12, kyy = o % 12;
    int kxv = (kx_i < 12) ? kx_i : (168 + kx_i);
    float re = 0.f, im = 0.f;
    int t = 0;
    for (int xx = 0; xx < 192; ++xx) {
      float2 v = gl[kyy * 192 + xx];
      float2 tt = tw[t];
      re += v.x * tt.x - v.y * tt.y;
      im += v.x * tt.y + v.y * tt.x;
      t += kxv;
      if (t >= 192) t -= 192;
    }
    ((float2*)S)[(size_t)bc * 288 + o] = make_float2(re, im);
  }
}

// ---------------------------------------------------------------------------
// Kernel 3: U[b,f,mode] = sum_t S[b,t,mode] * km[t,f,mode]   (complex).
// One block per mode (288 blocks).
// ---------------------------------------------------------------------------
__global__ __launch_bounds__(256) void k_specmul(const float* __restrict__ Sf,
                                                 const float* __restrict__ kmr,
                                                 const float* __restrict__ kmi,
                                                 float* __restrict__ Uf) {
  __shared__ float2 Sl[1024];  // [b][t]
  __shared__ float2 Kl[1024];  // [t][f]
  const int m = blockIdx.x;
  const int kx_i = m / 12, kyy = m % 12;
  const int kmode = (kx_i % 12) * 12 + kyy;
  const int tid = threadIdx.x;
  const float2* Sp = (const float2*)Sf;

  for (int i = tid; i < 1024; i += 256) {
    Sl[i] = Sp[(size_t)i * 288 + m];  // i = b*32 + t
    int t = i >> 5, f = i & 31;
    Kl[(t << 5) | f] = make_float2(kmr[(t * 32 + f) * 144 + kmode],
                                   kmi[(t * 32 + f) * 144 + kmode]);
  }
  __syncthreads();

  for (int o = tid; o < 1024; o += 256) {
    int b = o >> 5, f = o & 31;
    float re = 0.f, im = 0.f;
#pragma unroll 8
    for (int t = 0; t < 32; ++t) {
      float2 s = Sl[(b << 5) | t];
      float2 k = Kl[(t << 5) | f];
      re += s.x * k.x - s.y * k.y;
      im += s.x * k.y + s.y * k.x;
    }
    ((float2*)Uf)[(size_t)o * 288 + m] = make_float2(re, im);
  }
}

// ---------------------------------------------------------------------------
// Kernel 4: A += S (or A = S if init);  S += T(U).
// T = identity except ky==0 column: Hermitian symmetrization over kx.
// ---------------------------------------------------------------------------
__global__ __launch_bounds__(256) void k_specupd(float* __restrict__ Sf,
                                                 const float* __restrict__ Uf,
                                                 float* __restrict__ Af,
                                                 int init) {
  const int e = blockIdx.x * 256 + threadIdx.x;
  float2* S = (float2*)Sf;
  const float2* U = (const float2*)Uf;
  float2* A = (float2*)Af;
  const int m = e % 288, bt = e / 288;
  const int kx_i = m / 12, kyy = m % 12;

  float2 s = S[e];
  float2 u = U[e];
  float2 tu;
  if (kyy != 0) {
    tu = u;
  } else if (kx_i == 0) {
    tu = make_float2(u.x, 0.f);  // Re only at (0,0)
  } else if (kx_i == 12) {
    tu = make_float2(0.5f * u.x, 0.5f * u.y);  // partner kx=12 not retained
  } else {
    float2 up = U[bt * 288 + (24 - kx_i) * 12];  // kx <-> 192-kx
    tu = make_float2(0.5f * (u.x + up.x), 0.5f * (u.y - up.y));
  }
  float2 a = init ? s : make_float2(A[e].x + s.x, A[e].y + s.y);
  A[e] = a;
  S[e] = make_float2(s.x + tu.x, s.y + tu.y);
}

// ---------------------------------------------------------------------------
// Kernel 5: inverse DFT along x (with 1/192): z[b,ch,x,ky] from W[b,ch,mode].
// One block per (b,ch).
// ---------------------------------------------------------------------------
__global__ __launch_bounds__(256) void k_idftx(const float* __restrict__ Wf,
                                               float* __restrict__ zf) {
  __shared__ float2 Wl[288];
  __shared__ float2 tw[192];
  const int bc = blockIdx.x;
  const int tid = threadIdx.x;
  for (int i = tid; i < 192; i += 256) {
    float s, c;
    sincosf(TWO_PI_OVER_N * (float)i, &s, &c);
    tw[i] = make_float2(c, s);  // e^{+2*pi*i*t/192}
  }
  const float2* Wp = (const float2*)Wf + (size_t)bc * 288;
  for (int i = tid; i < 288; i += 256) Wl[i] = Wp[i];
  __syncthreads();

  const float inv = 1.0f / 192.0f;
  for (int o = tid; o < 2304; o += 256) {
    int xx = o / 12, kyy = o % 12;
    float re = 0.f, im = 0.f;
    int t1 = 0;                              // (kx*xx)%192 for kx = 0..11
    int t2 = (192 - ((12 * xx) % 192)) % 192;  // for kx = 180..191
#pragma unroll
    for (int j = 0; j < 12; ++j) {
      float2 w = Wl[j * 12 + kyy];
      float2 tt = tw[t1];
      re += w.x * tt.x - w.y * tt.y;
      im += w.x * tt.y + w.y * tt.x;
      float2 w2 = Wl[(j + 12) * 12 + kyy];
      float2 t2t = tw[t2];
      re += w2.x * t2t.x - w2.y * t2t.y;
      im += w2.x * t2t.y + w2.y * t2t.x;
      t1 += xx;
      if (t1 >= 192) t1 -= 192;
      t2 += xx;
      if (t2 >= 192) t2 -= 192;
    }
    ((float2*)zf)[((size_t)bc * 192 + xx) * 12 + kyy] =
        make_float2(re * inv, im * inv);
  }
}

// ---------------------------------------------------------------------------
// Kernel 6: finale. Per (b, x-row):
//   x_w = tanh(enc(x));  d = irfft_y(z) (K=24, WMMA, B from global table);
//   acc += w0 * x_w (K=32, WMMA);  x_raw = acc + w0_b + x_w -> d_out[part2];
//   out = dec * tanh(x_raw) + dec_b -> d_out[part1]
// ---------------------------------------------------------------------------
__global__ __launch_bounds__(128) void k_final(
    const float* __restrict__ x, const float* __restrict__ enc_w,
    const float* __restrict__ enc_b, const float* __restrict__ w0_w,
    const float* __restrict__ w0_b, const float* __restrict__ dec_w,
    const float* __restrict__ dec_b, const float* __restrict__ zf,
    const float* __restrict__ B2g, float* __restrict__ out) {
  __shared__ float xin[6][193];
  __shared__ float xw[32][193];
  __shared__ float xt[32][193];
  __shared__ float zz[32][25];  // interleaved re/im per ky, padded
  __shared__ float ws0[32][33];
  __shared__ float wenc[192];
  __shared__ float benc[32];
  __shared__ float wd[128];
  __shared__ float bd[4];
  __shared__ float b0[32];

  const int bid = blockIdx.x;
  const int b = bid / 192, xr = bid % 192;
  const int tid = threadIdx.x;
  const size_t XRAW_OFF = (size_t)32 * 4 * 192 * 192;

  for (int i = tid; i < 192; i += 128) wenc[i] = enc_w[i];
  if (tid < 32) { benc[tid] = enc_b[tid]; b0[tid] = w0_b[tid]; }
  wd[tid] = dec_w[tid];  // 128 threads, 128 weights
  if (tid < 4) bd[tid] = dec_b[tid];
  for (int i = tid; i < 1024; i += 128) ws0[i >> 5][i & 31] = w0_w[i];
  for (int i = tid; i < 6 * 192; i += 128) {
    int c = i / 192, y = i % 192;
    xin[c][y] = x[((b * 6 + c) * 192 + xr) * 192 + y];
  }
  const float2* zp = (const float2*)zf;
  for (int i = tid; i < 384; i += 128) {
    int ch = i / 12, kyy = i % 12;
    float2 v = zp[((size_t)(b * 32 + ch) * 192 + xr) * 12 + kyy];
    zz[ch][2 * kyy] = v.x;
    zz[ch][2 * kyy + 1] = v.y;
  }
  __syncthreads();

  for (int i = tid; i < 32 * 192; i += 128) {
    int ch = i / 192, y = i % 192;
    float s = benc[ch];
#pragma unroll
    for (int c = 0; c < 6; ++c) s += wenc[ch * 6 + c] * xin[c][y];
    xw[ch][y] = tanhf(s);
  }
  __syncthreads();

  const int lane = tid & 31, wv = tid >> 5;
  const int mtile = (wv >> 1) * 16;  // output-channel tile
  const int ntbase = (wv & 1) * 6;   // 6 column-tiles of 16 per wave
  const int h = lane >> 4, l15 = lane & 15;

  for (int nt = 0; nt < 6; ++nt) {
    const int col = (ntbase + nt) * 16 + l15;  // y coordinate
    v8f acc = {0.f, 0.f, 0.f, 0.f, 0.f, 0.f, 0.f, 0.f};

    // d = irfft along y (K=24); B fragments from L2-resident global table
    for (int k0 = 0; k0 < 24; k0 += 4) {
      v2f a, bf;
      a.x = zz[mtile + l15][k0 + 2 * h];
      a.y = zz[mtile + l15][k0 + 2 * h + 1];
      bf.x = B2g[(k0 + 2 * h) * 192 + col];
      bf.y = B2g[(k0 + 2 * h + 1) * 192 + col];
      acc = wmma_f32(a, bf, acc);
    }

    // acc += w0 * x_w  (K=32)
    for (int k0 = 0; k0 < 32; k0 += 4) {
      v2f a, bf;
      a.x = ws0[mtile + l15][k0 + 2 * h];
      a.y = ws0[mtile + l15][k0 + 2 * h + 1];
      bf.x = xw[k0 + 2 * h][col];
      bf.y = xw[k0 + 2 * h + 1][col];
      acc = wmma_f32(a, bf, acc);
    }

    // epilogue: x_raw = acc + w0_b + x_w ; write + tanh
#pragma unroll
    for (int i = 0; i < 8; ++i) {
      int f = mtile + 8 * h + i;
      float v = acc[i] + b0[f] + xw[f][col];
      out[XRAW_OFF + ((size_t)(b * 32 + f) * 192 + xr) * 192 + col] = v;
      xt[f][col] = tanhf(v);
    }
  }
  __syncthreads();

  // dec: out[b, co, xr, y]   (4 x 192 outputs, 32-deep dot)
  for (int o = tid; o < 768; o += 128) {
    int co = o / 192, y = o % 192;
    float s = bd[co];
#pragma unroll 8
    for (int ch = 0; ch < 32; ++ch) s += wd[co * 32 + ch] * xt[ch][y];
    out[((size_t)(b * 4 + co) * 192 + xr) * 192 + y] = s;
  }
}

// ---------------------------------------------------------------------------
// Launch.  Workspace layout (bytes):
//   g / z (aliased): 18,874,368   S: 2,359,296   A: 2,359,296
//   U: 2,359,296   W: 2,359,296   Fy: 18,432   B2: 18,432   (~28.4 MB)
// ---------------------------------------------------------------------------
extern "C" void kernel_launch(void* const* d_in, const int* in_sizes, int n_in,
                              void* d_out, int out_size, void* d_ws,
                              size_t ws_size, hipStream_t stream) {
  const float* x     = (const float*)d_in[0];
  const float* enc_w = (const float*)d_in[1];
  const float* enc_b = (const float*)d_in[2];
  const float* dec_w = (const float*)d_in[3];
  const float* dec_b = (const float*)d_in[4];
  const float* w0_w  = (const float*)d_in[5];
  const float* w0_b  = (const float*)d_in[6];
  const float* km_r  = (const float*)d_in[7];
  const float* km_i  = (const float*)d_in[8];
  float* outp = (float*)d_out;

  char* ws = (char*)d_ws;
  float* g   = (float*)(ws + 0);  // also reused as z
  float* S   = (float*)(ws + 18874368);
  float* A   = (float*)(ws + 21233664);
  float* U   = (float*)(ws + 23592960);
  float* W   = (float*)(ws + 25952256);
  float* Fyg = (float*)(ws + 28311552);
  float* B2g = (float*)(ws + 28329984);
  float* z = g;

  k_tables<<<18, 256, 0, stream>>>(Fyg, B2g);
  k_enc_dfty<<<32 * 192, 128, 0, stream>>>(x, enc_w, enc_b, Fyg, g);
  k_dftx<<<32 * 32, 256, 0, stream>>>(g, S);
  for (int j = 0; j < 6; ++j) {
    k_specmul<<<288, 256, 0, stream>>>(S, km_r, km_i, U);
    k_specupd<<<1152, 256, 0, stream>>>(S, U, A, (j == 0) ? 1 : 0);
  }
  k_specmul<<<288, 256, 0, stream>>>(A, km_r, km_i, W);
  k_idftx<<<32 * 32, 256, 0, stream>>>(W, z);
  k_final<<<32 * 192, 128, 0, stream>>>(x, enc_w, enc_b, w0_w, w0_b, dec_w,
                                        dec_b, z, B2g, outp);
}